// MCLA_douban_Encoder_2284922602170
// MI455X (gfx1250) — compile-verified
//
#include <hip/hip_runtime.h>
#include <math.h>

// ---- problem constants (match reference) ----
#define U_NODES 60000
#define I_NODES 90000
#define N_NODES 150000
#define DIM     64
#define NNZR    1000000
#define NNZS    1000000
#define CR      5   // N_RATINGS

typedef __attribute__((ext_vector_type(2))) float v2f;
typedef __attribute__((ext_vector_type(8))) float v8f;

// =====================================================================
// SpMM (COO scatter-add): out[rows[e]] += vals[e] * x[cols[e]]
// One wave (32 lanes) per edge; each lane handles a float2 of the D=64 row.
// x is split at `split`: col < split -> xa, else xb (bipartite concat).
// =====================================================================
__global__ void spmm_kernel(const int* __restrict__ rows, const int* __restrict__ cols,
                            const float* __restrict__ vals,
                            const float* __restrict__ xa, const float* __restrict__ xb,
                            int split, float* __restrict__ out, int nnz) {
  int e    = (blockIdx.x * blockDim.x + threadIdx.x) >> 5;
  int lane = threadIdx.x & 31;
  if (e >= nnz) return;
  int   r = rows[e];
  int   c = cols[e];
  float v = vals[e];
  const float* x = (c < split) ? (xa + (size_t)c * DIM)
                               : (xb + (size_t)(c - split) * DIM);
  float* o = out + (size_t)r * DIM;
  float2 xv = *(const float2*)(x + lane * 2);
  unsafeAtomicAdd(&o[lane * 2 + 0], v * xv.x);   // -> global_atomic_add_f32
  unsafeAtomicAdd(&o[lane * 2 + 1], v * xv.y);
}

// =====================================================================
// Elementwise mean of two layers: a = (a + b) * 0.5   (float4 granularity)
// =====================================================================
__global__ void avg_kernel(float* __restrict__ a, const float* __restrict__ b, int n4) {
  int i = blockIdx.x * blockDim.x + threadIdx.x;
  if (i >= n4) return;
  float4 x = ((float4*)a)[i];
  float4 y = ((const float4*)b)[i];
  x.x = (x.x + y.x) * 0.5f; x.y = (x.y + y.y) * 0.5f;
  x.z = (x.z + y.z) * 0.5f; x.w = (x.w + y.w) * 0.5f;
  ((float4*)a)[i] = x;
}

// =====================================================================
// Row-wise L2 normalize, clip 1e-12. One wave per row of 64 floats.
// =====================================================================
__global__ void l2norm_kernel(float* __restrict__ x, int nrows) {
  int row  = (blockIdx.x * blockDim.x + threadIdx.x) >> 5;
  int lane = threadIdx.x & 31;
  if (row >= nrows) return;
  float2 v = *(float2*)(x + (size_t)row * DIM + lane * 2);
  float s = v.x * v.x + v.y * v.y;
  #pragma unroll
  for (int off = 16; off > 0; off >>= 1) s += __shfl_xor(s, off);
  float inv = 1.0f / fmaxf(sqrtf(s), 1e-12f);
  v.x *= inv; v.y *= inv;
  *(float2*)(x + (size_t)row * DIM + lane * 2) = v;
}

// =====================================================================
// Fused attention channel mixer, WMMA f32 16x16x4.
//   w[c,n]   = tanh( (emb[c,n,:] @ M^T) @ a )   (exact unfolded form, f32 WMMA)
//   score    = softmax_c(w)                      -> scoresOut [C, numNodes]
//   mixed[n] = sum_c score[c,n] * emb[c,n,:]     -> mixedOut  [numNodes, 64]
// One wave per 16-node tile. Emb tiles + M cached in LDS (rows padded to 68
// floats to avoid 16-way bank conflicts on the stride-64 fragment reads).
// numNodes must be a multiple of 16 (U=60000, I=90000 both are) so EXEC is
// all-ones at every WMMA (ISA requirement).
// =====================================================================
template<int C>
__global__ void __launch_bounds__(32)
att_fuse_kernel(const float* __restrict__ embBase, size_t chanStride,
                const float* __restrict__ Mmat, const float* __restrict__ avec,
                float* __restrict__ scoresOut, float* __restrict__ mixedOut,
                int numNodes) {
  __shared__ __align__(16) float Ms[64 * 68];        // M row-major, padded
  __shared__ __align__(16) float embS[C][16 * 68];   // per-channel 16x64 tile
  __shared__ float wS[C][16];

  const int lane  = threadIdx.x;
  const int node0 = blockIdx.x * 16;

  // cache M (64x64) into LDS with row pad 68
  for (int i = lane; i < 4096; i += 32) {
    int e = i >> 6, d = i & 63;
    Ms[e * 68 + d] = Mmat[i];
  }
  // cache emb tiles (C x 16 x 64)
  for (int c = 0; c < C; ++c) {
    const float* src = embBase + (size_t)c * chanStride + (size_t)node0 * DIM;
    for (int i = lane; i < 1024; i += 32) {
      int r = i >> 6, d = i & 63;
      embS[c][r * 68 + d] = src[i];
    }
  }
  __syncthreads();

  const int half = lane >> 4;     // 0: lanes 0-15, 1: lanes 16-31
  const int l16  = lane & 15;
  float aL[4];
  #pragma unroll
  for (int t = 0; t < 4; ++t) aL[t] = avec[t * 16 + l16];

  for (int c = 0; c < C; ++c) {
    // 4 C-tiles cover output cols e = 0..63 for 16 nodes
    v8f acc0 = {}, acc1 = {}, acc2 = {}, acc3 = {};
    #pragma unroll
    for (int k = 0; k < 64; k += 4) {
      // A frag (16x4 f32): lanes 0-15 -> K=k..k+1, lanes 16-31 -> K=k+2..k+3
      v2f A;
      A.x = embS[c][l16 * 68 + k + half * 2 + 0];
      A.y = embS[c][l16 * 68 + k + half * 2 + 1];
      // B frag (4x16 f32) of M^T: B[kk][e] = M[e][kk]
      v2f B0, B1, B2, B3;
      B0.x = Ms[(0 * 16 + l16) * 68 + k + half * 2 + 0];
      B0.y = Ms[(0 * 16 + l16) * 68 + k + half * 2 + 1];
      B1.x = Ms[(1 * 16 + l16) * 68 + k + half * 2 + 0];
      B1.y = Ms[(1 * 16 + l16) * 68 + k + half * 2 + 1];
      B2.x = Ms[(2 * 16 + l16) * 68 + k + half * 2 + 0];
      B2.y = Ms[(2 * 16 + l16) * 68 + k + half * 2 + 1];
      B3.x = Ms[(3 * 16 + l16) * 68 + k + half * 2 + 0];
      B3.y = Ms[(3 * 16 + l16) * 68 + k + half * 2 + 1];
      acc0 = __builtin_amdgcn_wmma_f32_16x16x4_f32(false, A, false, B0, (short)0, acc0, false, false);
      acc1 = __builtin_amdgcn_wmma_f32_16x16x4_f32(false, A, false, B1, (short)0, acc1, false, false);
      acc2 = __builtin_amdgcn_wmma_f32_16x16x4_f32(false, A, false, B2, (short)0, acc2, false, false);
      acc3 = __builtin_amdgcn_wmma_f32_16x16x4_f32(false, A, false, B3, (short)0, acc3, false, false);
    }
    // C-tile layout: VGPR r, lane -> out[m = r + half*8][e = t*16 + l16]
    // w[m] = sum_e out[m][e] * a[e]; reduce across the 16 lanes of each half.
    #pragma unroll
    for (int r = 0; r < 8; ++r) {
      float sv = acc0[r] * aL[0] + acc1[r] * aL[1] + acc2[r] * aL[2] + acc3[r] * aL[3];
      sv += __shfl_xor(sv, 1);
      sv += __shfl_xor(sv, 2);
      sv += __shfl_xor(sv, 4);
      sv += __shfl_xor(sv, 8);
      if (l16 == 0) wS[c][half * 8 + r] = tanhf(sv);
    }
  }
  __syncthreads();

  // softmax over channels, emit scores
  if (lane < 16) {
    float mx = -3.4e38f;
    for (int c = 0; c < C; ++c) mx = fmaxf(mx, wS[c][lane]);
    float ex[C];
    float sum = 0.0f;
    for (int c = 0; c < C; ++c) { ex[c] = expf(wS[c][lane] - mx); sum += ex[c]; }
    float inv = 1.0f / sum;
    for (int c = 0; c < C; ++c) {
      float sc = ex[c] * inv;
      wS[c][lane] = sc;
      scoresOut[(size_t)c * numNodes + node0 + lane] = sc;
    }
  }
  __syncthreads();

  // mixed[n] = sum_c score * emb   (each lane owns a float2 of each row)
  #pragma unroll
  for (int m = 0; m < 16; ++m) {
    float2 acc; acc.x = 0.0f; acc.y = 0.0f;
    for (int c = 0; c < C; ++c) {
      float sc = wS[c][m];
      float2 ev = *(const float2*)&embS[c][m * 68 + lane * 2];
      acc.x += sc * ev.x; acc.y += sc * ev.y;
    }
    *(float2*)&mixedOut[(size_t)(node0 + m) * DIM + lane * 2] = acc;
  }
}

// =====================================================================
// Adaptive combine: per user row
//   sim = clamp cos(mru, ms); rate = 1/(2 + max(sim,0)*adapt); out = mru + ms*rate
// One wave per row.
// =====================================================================
__global__ void combine_kernel(const float* __restrict__ mru, const float* __restrict__ ms,
                               const float* __restrict__ adapt,
                               float* __restrict__ outMixed, int nU) {
  int idx  = blockIdx.x * blockDim.x + threadIdx.x;
  int row  = idx >> 5;
  int lane = idx & 31;
  if (row >= nU) return;
  float2 u = *(const float2*)&mru[(size_t)row * DIM + lane * 2];
  float2 s = *(const float2*)&ms [(size_t)row * DIM + lane * 2];
  float d  = u.x * s.x + u.y * s.y;
  float nu = u.x * u.x + u.y * u.y;
  float ns = s.x * s.x + s.y * s.y;
  #pragma unroll
  for (int off = 16; off > 0; off >>= 1) {
    d  += __shfl_xor(d, off);
    nu += __shfl_xor(nu, off);
    ns += __shfl_xor(ns, off);
  }
  float denom = fmaxf(sqrtf(nu), 1e-8f) * fmaxf(sqrtf(ns), 1e-8f);
  float sim   = d / denom;
  float rate  = 1.0f / (2.0f + fmaxf(sim, 0.0f) * adapt[row]);   // base=(2+2)/2, EMB_COMB=1
  float2 o; o.x = u.x + s.x * rate; o.y = u.y + s.y * rate;
  *(float2*)&outMixed[(size_t)row * DIM + lane * 2] = o;
}

// =====================================================================
extern "C" void kernel_launch(void* const* d_in, const int* in_sizes, int n_in,
                              void* d_out, int out_size, void* d_ws, size_t ws_size,
                              hipStream_t stream) {
  (void)in_sizes; (void)n_in; (void)out_size; (void)ws_size;
  const int*   r_rows  = (const int*)  d_in[0];
  const int*   r_cols  = (const int*)  d_in[1];
  const float* r_vals  = (const float*)d_in[2];
  const int*   f_rows  = (const int*)  d_in[3];
  const int*   f_cols  = (const int*)  d_in[4];
  const float* f_vals  = (const float*)d_in[5];
  const int*   g_rows  = (const int*)  d_in[6];
  const int*   g_cols  = (const int*)  d_in[7];
  const float* g_vals  = (const float*)d_in[8];
  const float* ruemb   = (const float*)d_in[9];
  const float* riemb   = (const float*)d_in[10];
  const float* femb    = (const float*)d_in[11];
  const float* gemb    = (const float*)d_in[12];
  const float* u_att_w = (const float*)d_in[13];
  const float* u_att_M = (const float*)d_in[14];
  const float* i_att_w = (const float*)d_in[15];
  const float* i_att_M = (const float*)d_in[16];
  const float* s_att_w = (const float*)d_in[17];
  const float* s_att_M = (const float*)d_in[18];
  const float* adapt   = (const float*)d_in[19];

  // ---- workspace layout (floats) ----
  float* ws = (float*)d_ws;
  const size_t SZ_ND = (size_t)N_NODES * DIM;   // 9.6M
  const size_t SZ_UD = (size_t)U_NODES * DIM;   // 3.84M
  float* mean_embs = ws;                                 // [5, N, 64]
  float* tmp       = mean_embs + (size_t)CR * SZ_ND;     // [N, 64]
  float* F1        = tmp + SZ_ND;                        // [U, 64]
  float* G1        = F1 + SZ_UD;                         // [U, 64] (contiguous with F1)
  float* mru       = G1 + SZ_UD;                         // mixed_rating_user [U,64]
  float* msoc      = mru + SZ_UD;                        // mixed_social      [U,64]

  // ---- output layout (floats, reference return order) ----
  float* out           = (float*)d_out;
  float* o_user_scores = out;                                    // [5, U]
  float* o_item_scores = o_user_scores + (size_t)CR * U_NODES;   // [5, I]
  float* o_soc_scores  = o_item_scores + (size_t)CR * I_NODES;   // [2, U]
  float* o_mixed_user  = o_soc_scores + (size_t)2 * U_NODES;     // [U, 64]
  float* o_mixed_item  = o_mixed_user + SZ_UD;                   // [I, 64]

  const int SPMM_GRID_R = NNZR / 8;   // 8 edges per 256-thread block
  const int SPMM_GRID_S = NNZS / 8;

  // ---- rating channels: 2-layer LightGCN propagation, mean of layers ----
  for (int c = 0; c < CR; ++c) {
    float* mc = mean_embs + (size_t)c * SZ_ND;
    const int* rr = r_rows + (size_t)c * NNZR;
    const int* rc = r_cols + (size_t)c * NNZR;
    const float* rv = r_vals + (size_t)c * NNZR;
    hipMemsetAsync(mc, 0, SZ_ND * sizeof(float), stream);
    spmm_kernel<<<SPMM_GRID_R, 256, 0, stream>>>(rr, rc, rv,
        ruemb + (size_t)c * U_NODES * DIM, riemb + (size_t)c * I_NODES * DIM,
        U_NODES, mc, NNZR);                       // layer 1 -> mc
    hipMemsetAsync(tmp, 0, SZ_ND * sizeof(float), stream);
    spmm_kernel<<<SPMM_GRID_R, 256, 0, stream>>>(rr, rc, rv,
        mc, mc + SZ_UD, U_NODES, tmp, NNZR);      // layer 2 -> tmp
    avg_kernel<<<(int)(SZ_ND / 4 + 255) / 256, 256, 0, stream>>>(mc, tmp, (int)(SZ_ND / 4));
  }

  // ---- social channels: spmm + per-layer L2 norm, mean of layers ----
  // friend
  hipMemsetAsync(F1, 0, SZ_UD * sizeof(float), stream);
  spmm_kernel<<<SPMM_GRID_S, 256, 0, stream>>>(f_rows, f_cols, f_vals, femb, femb, U_NODES, F1, NNZS);
  l2norm_kernel<<<(U_NODES + 7) / 8, 256, 0, stream>>>(F1, U_NODES);
  hipMemsetAsync(tmp, 0, SZ_UD * sizeof(float), stream);
  spmm_kernel<<<SPMM_GRID_S, 256, 0, stream>>>(f_rows, f_cols, f_vals, F1, F1, U_NODES, tmp, NNZS);
  l2norm_kernel<<<(U_NODES + 7) / 8, 256, 0, stream>>>(tmp, U_NODES);
  avg_kernel<<<(int)(SZ_UD / 4 + 255) / 256, 256, 0, stream>>>(F1, tmp, (int)(SZ_UD / 4));
  // group
  hipMemsetAsync(G1, 0, SZ_UD * sizeof(float), stream);
  spmm_kernel<<<SPMM_GRID_S, 256, 0, stream>>>(g_rows, g_cols, g_vals, gemb, gemb, U_NODES, G1, NNZS);
  l2norm_kernel<<<(U_NODES + 7) / 8, 256, 0, stream>>>(G1, U_NODES);
  hipMemsetAsync(tmp, 0, SZ_UD * sizeof(float), stream);
  spmm_kernel<<<SPMM_GRID_S, 256, 0, stream>>>(g_rows, g_cols, g_vals, G1, G1, U_NODES, tmp, NNZS);
  l2norm_kernel<<<(U_NODES + 7) / 8, 256, 0, stream>>>(tmp, U_NODES);
  avg_kernel<<<(int)(SZ_UD / 4 + 255) / 256, 256, 0, stream>>>(G1, tmp, (int)(SZ_UD / 4));

  // ---- attention fusion (WMMA f32 16x16x4) ----
  att_fuse_kernel<CR><<<U_NODES / 16, 32, 0, stream>>>(
      mean_embs, SZ_ND, u_att_M, u_att_w, o_user_scores, mru, U_NODES);
  att_fuse_kernel<CR><<<I_NODES / 16, 32, 0, stream>>>(
      mean_embs + SZ_UD, SZ_ND, i_att_M, i_att_w, o_item_scores, o_mixed_item, I_NODES);
  att_fuse_kernel<2><<<U_NODES / 16, 32, 0, stream>>>(
      F1, SZ_UD, s_att_M, s_att_w, o_soc_scores, msoc, U_NODES);

  // ---- adaptive combine ----
  combine_kernel<<<(U_NODES * 32 + 255) / 256, 256, 0, stream>>>(
      mru, msoc, adapt, o_mixed_user, U_NODES);
}